// SoftMoE_FFN_58231166599595
// MI455X (gfx1250) — compile-verified
//
#include <hip/hip_runtime.h>
#include <hip/hip_bf16.h>

// Problem constants (match the reference)
#define B_  8
#define T_  2048
#define D_  1024
#define E_  8        // E * S slots
#define FF_ 4096

typedef __attribute__((ext_vector_type(2))) float v2f;
typedef __attribute__((ext_vector_type(8))) float v8f;

__device__ __forceinline__ v8f wmma_f32_16x16x4(v2f a, v2f b, v8f c) {
    // 8 args: (neg_a, A, neg_b, B, c_mod, C, reuse_a, reuse_b)
    return __builtin_amdgcn_wmma_f32_16x16x4_f32(
        false, a, false, b, (short)0, c, false, false);
}

// ---------------------------------------------------------------------------
// Kernel 1: logits[b,t,e] = X[b,t,:] . Wg[:,e] + bg[e]
// 16 tokens per wave-tile; N padded 8 -> 16 by DUPLICATING Wg columns (r&7):
// garbage output columns 8..15 are simply never stored, so no predication
// is needed in the inner loop (branch-free, EXEC stays all-1s).
// ---------------------------------------------------------------------------
__global__ void k_logits(const float* __restrict__ X, const float* __restrict__ Wg,
                         const float* __restrict__ bg, float* __restrict__ logits) {
    const int wid  = threadIdx.x >> 5;
    const int lane = threadIdx.x & 31;
    const int g    = blockIdx.x * (blockDim.x >> 5) + wid;   // tile over B*T/16 rows
    const int r    = lane & 15;
    const int c    = lane & 7;          // clamped (duplicated) column for Wg
    const int half = lane >> 4;
    const long row0 = (long)g * 16;

    const float* xrow = X + (row0 + r) * D_;   // A-matrix: lane r = row, streams K
    v8f acc = {};
    #pragma unroll 4
    for (int k = 0; k < D_; k += 4) {
        const int k0 = k + half * 2;
        v2f a, bmat;
        a.x = xrow[k0];
        a.y = xrow[k0 + 1];
        bmat.x = Wg[(long)k0 * E_ + c];
        bmat.y = Wg[(long)(k0 + 1) * E_ + c];
        acc = wmma_f32_16x16x4(a, bmat, acc);
    }
    if (r < E_) {
        const float bias = bg[r];
        #pragma unroll
        for (int i = 0; i < 8; ++i) {
            const long m = row0 + i + half * 8;   // C layout: VGPR i, halves are M / M+8
            logits[m * E_ + r] = acc[i] + bias;
        }
    }
}

// ---------------------------------------------------------------------------
// Kernel 2: dispatch_w = softmax over tokens T (axis=1), per (b, e)
// ---------------------------------------------------------------------------
__global__ void k_dispatch_softmax(const float* __restrict__ logits,
                                   float* __restrict__ disp) {
    __shared__ float red[256];
    const int b   = blockIdx.x >> 3;
    const int e   = blockIdx.x & 7;
    const int tid = threadIdx.x;
    const float* base = logits + (long)b * T_ * E_ + e;

    float lmax = -1e30f;
    for (int t = tid; t < T_; t += 256) lmax = fmaxf(lmax, base[(long)t * E_]);
    red[tid] = lmax; __syncthreads();
    for (int s = 128; s > 0; s >>= 1) {
        if (tid < s) red[tid] = fmaxf(red[tid], red[tid + s]);
        __syncthreads();
    }
    const float m = red[0]; __syncthreads();

    float lsum = 0.f;
    for (int t = tid; t < T_; t += 256) lsum += __expf(base[(long)t * E_] - m);
    red[tid] = lsum; __syncthreads();
    for (int s = 128; s > 0; s >>= 1) {
        if (tid < s) red[tid] += red[tid + s];
        __syncthreads();
    }
    const float inv = 1.f / red[0];

    float* dbase = disp + (long)b * T_ * E_ + e;
    for (int t = tid; t < T_; t += 256)
        dbase[(long)t * E_] = __expf(base[(long)t * E_] - m) * inv;
}

// ---------------------------------------------------------------------------
// Kernel 3: combine_w = softmax over the 8 slots (axis=2), per token
// ---------------------------------------------------------------------------
__global__ void k_combine_softmax(const float* __restrict__ logits,
                                  float* __restrict__ comb) {
    const long t = (long)blockIdx.x * blockDim.x + threadIdx.x;   // over B*T
    const float* p = logits + t * E_;
    float v[E_];
    float m = -1e30f;
    #pragma unroll
    for (int e = 0; e < E_; ++e) { v[e] = p[e]; m = fmaxf(m, v[e]); }
    float s = 0.f;
    #pragma unroll
    for (int e = 0; e < E_; ++e) { v[e] = __expf(v[e] - m); s += v[e]; }
    const float inv = 1.f / s;
    float* q = comb + t * E_;
    #pragma unroll
    for (int e = 0; e < E_; ++e) q[e] = v[e] * inv;
}

// ---------------------------------------------------------------------------
// Kernel 4: slots[b,e,d] = sum_t disp[b,t,e] * X[b,t,d]
// Per batch: (8 x 2048) @ (2048 x 1024). A rows 8..15 are DUPLICATES (r&7) of
// valid rows -> garbage output rows 8..15 are never stored; inner loop is
// branch-free. X is the B-matrix -> lanes read consecutive d (coalesced).
// ---------------------------------------------------------------------------
__global__ void k_slots(const float* __restrict__ disp, const float* __restrict__ X,
                        float* __restrict__ slots) {
    const int wid  = threadIdx.x >> 5;
    const int lane = threadIdx.x & 31;
    const int g    = blockIdx.x * (blockDim.x >> 5) + wid;   // 512 tiles: b*(D/16)+nt
    const int b    = g >> 6;
    const int n0   = (g & 63) << 4;
    const int r    = lane & 15;
    const int c    = lane & 7;          // clamped (duplicated) expert row
    const int half = lane >> 4;

    const float* dbase = disp + (long)b * T_ * E_;
    const float* xbase = X    + (long)b * T_ * D_;
    v8f acc = {};
    #pragma unroll 4
    for (int k = 0; k < T_; k += 4) {
        const int k0 = k + half * 2;
        v2f a, bmat;
        a.x = dbase[(long)k0 * E_ + c];          // A^T: row e, col t
        a.y = dbase[(long)(k0 + 1) * E_ + c];
        bmat.x = xbase[(long)k0 * D_ + n0 + r];
        bmat.y = xbase[(long)(k0 + 1) * D_ + n0 + r];
        acc = wmma_f32_16x16x4(a, bmat, acc);
    }
    if (half == 0) {   // rows 8..15 are padding; lanes 0-15 hold M=0..7
        #pragma unroll
        for (int i = 0; i < 8; ++i)
            slots[((long)b * E_ + i) * D_ + n0 + r] = acc[i];
    }
}

// ---------------------------------------------------------------------------
// Kernel 5: H = silu(slots @ W1 + b1); slots is (64 x 1024), W1 (1024 x 4096)
// ---------------------------------------------------------------------------
__global__ void k_ffn1(const float* __restrict__ slots, const float* __restrict__ W1,
                       const float* __restrict__ b1, float* __restrict__ H) {
    const int wid  = threadIdx.x >> 5;
    const int lane = threadIdx.x & 31;
    const int g    = blockIdx.x * (blockDim.x >> 5) + wid;   // 4 Mtiles * 256 Ntiles
    const int mt   = g >> 8;
    const int n0   = (g & 255) << 4;
    const int r    = lane & 15;
    const int half = lane >> 4;

    const float* arow = slots + (long)(mt * 16 + r) * D_;
    v8f acc = {};
    #pragma unroll 4
    for (int k = 0; k < D_; k += 4) {
        const int k0 = k + half * 2;
        v2f a, bmat;
        a.x = arow[k0];
        a.y = arow[k0 + 1];
        bmat.x = W1[(long)k0 * FF_ + n0 + r];
        bmat.y = W1[(long)(k0 + 1) * FF_ + n0 + r];
        acc = wmma_f32_16x16x4(a, bmat, acc);
    }
    const float bias = b1[n0 + r];
    #pragma unroll
    for (int i = 0; i < 8; ++i) {
        const int m = mt * 16 + i + half * 8;
        const float v = acc[i] + bias;
        H[(long)m * FF_ + n0 + r] = v / (1.f + __expf(-v));   // SiLU
    }
}

// ---------------------------------------------------------------------------
// Kernel 6: Y = H @ W2 + b2; H is (64 x 4096), W2 (4096 x 1024)
// ---------------------------------------------------------------------------
__global__ void k_ffn2(const float* __restrict__ H, const float* __restrict__ W2,
                       const float* __restrict__ b2, float* __restrict__ Y) {
    const int wid  = threadIdx.x >> 5;
    const int lane = threadIdx.x & 31;
    const int g    = blockIdx.x * (blockDim.x >> 5) + wid;   // 4 Mtiles * 64 Ntiles
    const int mt   = g >> 6;
    const int n0   = (g & 63) << 4;
    const int r    = lane & 15;
    const int half = lane >> 4;

    const float* arow = H + (long)(mt * 16 + r) * FF_;
    v8f acc = {};
    #pragma unroll 4
    for (int k = 0; k < FF_; k += 4) {
        const int k0 = k + half * 2;
        v2f a, bmat;
        a.x = arow[k0];
        a.y = arow[k0 + 1];
        bmat.x = W2[(long)k0 * D_ + n0 + r];
        bmat.y = W2[(long)(k0 + 1) * D_ + n0 + r];
        acc = wmma_f32_16x16x4(a, bmat, acc);
    }
    const float bias = b2[n0 + r];
    #pragma unroll
    for (int i = 0; i < 8; ++i) {
        const int m = mt * 16 + i + half * 8;
        Y[(long)m * D_ + n0 + r] = acc[i] + bias;
    }
}

// ---------------------------------------------------------------------------
// Kernel 7: out[b,t,d] = sum_e comb[b,t,e] * Y[b,e,d]
// Stage Y[b] (8x1024 = 32KB) in LDS; 16 tokens per block; coalesced 64MB write.
// ---------------------------------------------------------------------------
__global__ void k_combine_out(const float* __restrict__ comb, const float* __restrict__ Y,
                              float* __restrict__ out) {
    __shared__ float yS[E_ * D_];      // 32 KB
    __shared__ float cwS[16 * E_];
    const int tid = threadIdx.x;
    const int b   = blockIdx.x >> 7;            // T/16 = 128 blocks per batch
    const int t0  = (blockIdx.x & 127) << 4;

    const float* ybase = Y + (long)b * E_ * D_;
    for (int j = tid; j < E_ * D_; j += 256) yS[j] = ybase[j];
    if (tid < 16 * E_) {
        const int tl = tid >> 3, e = tid & 7;
        cwS[tid] = comb[((long)b * T_ + t0 + tl) * E_ + e];
    }
    __syncthreads();

    float* obase = out + ((long)b * T_ + t0) * D_;
    for (int j = tid; j < 16 * D_; j += 256) {
        const int tl = j >> 10, d = j & 1023;
        float s = 0.f;
        #pragma unroll
        for (int e = 0; e < E_; ++e) s += cwS[tl * E_ + e] * yS[e * D_ + d];
        obase[(long)tl * D_ + d] = s;
    }
}

// ---------------------------------------------------------------------------
extern "C" void kernel_launch(void* const* d_in, const int* in_sizes, int n_in,
                              void* d_out, int out_size, void* d_ws, size_t ws_size,
                              hipStream_t stream) {
    const float* X  = (const float*)d_in[0];
    const float* Wg = (const float*)d_in[1];
    const float* bg = (const float*)d_in[2];
    const float* W1 = (const float*)d_in[3];
    const float* b1 = (const float*)d_in[4];
    const float* W2 = (const float*)d_in[5];
    const float* b2 = (const float*)d_in[6];
    float* out = (float*)d_out;

    // Workspace layout (floats): ~3 MB total
    float* ws     = (float*)d_ws;
    float* logits = ws;                          // B*T*E   = 131072
    float* disp   = logits + (long)B_ * T_ * E_; // 131072
    float* comb   = disp   + (long)B_ * T_ * E_; // 131072
    float* slots  = comb   + (long)B_ * T_ * E_; // B*E*D   = 65536
    float* H      = slots  + (long)B_ * E_ * D_; // B*E*FF  = 262144
    float* Y      = H      + (long)B_ * E_ * FF_;// B*E*D   = 65536

    // 1) logits = X @ Wg + bg     (WMMA; 1024 tiles / 8 waves per block)
    k_logits<<<dim3((B_ * T_ / 16) / 8), dim3(256), 0, stream>>>(X, Wg, bg, logits);
    // 2) dispatch softmax over T
    k_dispatch_softmax<<<dim3(B_ * E_), dim3(256), 0, stream>>>(logits, disp);
    // 3) combine softmax over slots
    k_combine_softmax<<<dim3(B_ * T_ / 256), dim3(256), 0, stream>>>(logits, comb);
    // 4) slots = dispatch^T @ X   (WMMA; 512 tiles)
    k_slots<<<dim3(64), dim3(256), 0, stream>>>(disp, X, slots);
    // 5) H = silu(slots @ W1 + b1) (WMMA; 1024 tiles)
    k_ffn1<<<dim3(128), dim3(256), 0, stream>>>(slots, W1, b1, H);
    // 6) Y = H @ W2 + b2           (WMMA; 256 tiles)
    k_ffn2<<<dim3(32), dim3(256), 0, stream>>>(H, W2, b2, Y);
    // 7) out = combine_w @ Y       (LDS-staged, coalesced output write)
    k_combine_out<<<dim3(B_ * 128), dim3(256), 0, stream>>>(comb, Y, out);
}